// BertHungarianLoss_5927054869180
// MI455X (gfx1250) — compile-verified
//
#include <hip/hip_runtime.h>
#include <hip/hip_bf16.h>

#define M_ROWS 10
#define VOCAB  50257
#define NPERM  3628800  // 10!
#define SCORE_BLOCK 256 // perms per block in the scoring kernel (NPERM % 256 == 0 -> 14175 blocks)

typedef __attribute__((ext_vector_type(2))) float v2f;
typedef __attribute__((ext_vector_type(8))) float v8f;

// ---------------------------------------------------------------------------
// Wave-level f32 sum using the CDNA5 f32 matrix pipe.
// A (16x4 f32): lane L<16 holds (M=L, K=0|1), lane L>=16 holds (M=L-16, K=2|3).
// We place each lane's partial in A.x (K=0 / K=2) and 0 in A.y.  With B = ones,
// C[m][n] = partial[m] + partial[m+16] for all n.  D layout: lane 0 VGPR i = C[i][0]
// (i=0..7), lane 16 VGPR i = C[i+8][0].  Two shuffles then give the full wave sum.
// Must be called with EXEC all-ones (uniform control flow).
// ---------------------------------------------------------------------------
__device__ __forceinline__ float wave_sum_f32_wmma(float partial) {
    v2f a; a.x = partial; a.y = 0.0f;
    v2f b; b.x = 1.0f;    b.y = 1.0f;
    v8f c = {};
    c = __builtin_amdgcn_wmma_f32_16x16x4_f32(
        /*neg_a=*/false, a, /*neg_b=*/false, b,
        /*c_mod=*/(short)0, c, /*reuse_a=*/false, /*reuse_b=*/false);
    float local = ((c[0] + c[1]) + (c[2] + c[3])) + ((c[4] + c[5]) + (c[6] + c[7]));
    // total = local@lane0 + local@lane16
    return __shfl(local, 0, 32) + __shfl(local, 16, 32);
}

// ---------------------------------------------------------------------------
// Kernel A: per-row max and logZ = max + log(sum(exp(x - max))).  One block/row.
// 2 MB of logits read twice -> bandwidth-trivial.
// ---------------------------------------------------------------------------
__global__ __launch_bounds__(256)
void k_rowstats(const float* __restrict__ logits, float* __restrict__ rowLogZ) {
    const int row  = blockIdx.x;
    const int tid  = threadIdx.x;
    const int lane = tid & 31;
    const int wid  = tid >> 5;
    const float* x = logits + (size_t)row * VOCAB;

    __shared__ float s_max[8];
    __shared__ float s_sum[8];
    __shared__ float s_rowmax;

    // pass 1: max
    float m = -INFINITY;
    for (int i = tid; i < VOCAB; i += 256) m = fmaxf(m, x[i]);
    #pragma unroll
    for (int o = 16; o > 0; o >>= 1) m = fmaxf(m, __shfl_xor(m, o, 32));
    if (lane == 0) s_max[wid] = m;
    __syncthreads();
    if (tid == 0) {
        float mm = s_max[0];
        #pragma unroll
        for (int i = 1; i < 8; ++i) mm = fmaxf(mm, s_max[i]);
        s_rowmax = mm;
    }
    __syncthreads();
    const float rmax = s_rowmax;

    // pass 2: sum of exp(x - max); wave reduction on the matrix pipe (v_wmma f32)
    float acc = 0.0f;
    for (int i = tid; i < VOCAB; i += 256) acc += __expf(x[i] - rmax);
    float wsum = wave_sum_f32_wmma(acc);
    if (lane == 0) s_sum[wid] = wsum;
    __syncthreads();
    if (tid == 0) {
        float t = 0.0f;
        #pragma unroll
        for (int i = 0; i < 8; ++i) t += s_sum[i];
        rowLogZ[row] = rmax + __logf(t);
    }
}

// ---------------------------------------------------------------------------
// Kernel B: S[j][c] = exp(logits[j, target[c]] - logZ_j)  (100 values),
// and reset the packed argmax accumulator (deterministic per launch).
// ---------------------------------------------------------------------------
__global__ void k_build_S(const float* __restrict__ logits,
                          const int* __restrict__ target,
                          const float* __restrict__ rowLogZ,
                          float* __restrict__ Sbuf,
                          unsigned long long* __restrict__ best) {
    const int t = threadIdx.x;
    if (t == 0) *best = 0ULL;
    if (t < 100) {
        const int j = t / 10, c = t % 10;
        const int col = target[c];
        Sbuf[t] = __expf(logits[(size_t)j * VOCAB + col] - rowLogZ[j]);
    }
}

// ---------------------------------------------------------------------------
// Kernel C: stream 145 MB of perms (the roofline term), score each permutation
// score[p] = sum_j S[j][perm[p][j]], and argmax with first-occurrence tie-break.
// Coalesced b32 loads into LDS tile; global_prefetch for the tile after next.
// ---------------------------------------------------------------------------
__global__ __launch_bounds__(256)
void k_score(const int* __restrict__ perms,
             const float* __restrict__ Sbuf,
             unsigned long long* __restrict__ best) {
    __shared__ float sS[100];
    __shared__ int   sp[SCORE_BLOCK * 10];
    __shared__ unsigned long long s_key[8];

    const int tid  = threadIdx.x;
    const int lane = tid & 31;
    const int wid  = tid >> 5;

    if (tid < 100) sS[tid] = Sbuf[tid];

    const unsigned base = (unsigned)blockIdx.x * SCORE_BLOCK;
    const int* src = perms + (size_t)base * 10;

    // prefetch the tile after next (cacheline-granular; lowers to global_prefetch_b8)
    __builtin_prefetch(src + 2 * SCORE_BLOCK * 10 + tid * 16, 0, 0);

    // coalesced load of 2560 dwords: sp[d] = src[d]
    #pragma unroll
    for (int k = 0; k < 10; ++k)
        sp[k * SCORE_BLOCK + tid] = src[k * SCORE_BLOCK + tid];
    __syncthreads();

    // score this thread's permutation (LDS: 10*tid stride -> conflict-free banks)
    float score = 0.0f;
    #pragma unroll
    for (int j = 0; j < 10; ++j)
        score += sS[j * 10 + sp[tid * 10 + j]];

    const unsigned idx = base + tid;
    // scores > 0 -> f32 bits are order-preserving as u32; ~idx gives min-index tie-break
    unsigned long long key =
        ((unsigned long long)__float_as_uint(score) << 32) |
        (unsigned long long)(0xFFFFFFFFu - idx);

    #pragma unroll
    for (int o = 16; o > 0; o >>= 1) {
        unsigned long long other = __shfl_xor(key, o, 32);
        key = (other > key) ? other : key;
    }
    if (lane == 0) s_key[wid] = key;
    __syncthreads();
    if (tid == 0) {
        unsigned long long k = s_key[0];
        #pragma unroll
        for (int i = 1; i < 8; ++i) k = (s_key[i] > k) ? s_key[i] : k;
        atomicMax(best, k);
    }
}

// ---------------------------------------------------------------------------
// Kernel D: decode best index, emit loss (out[0:10]) and target_best (out[10:20]).
// ---------------------------------------------------------------------------
__global__ void k_finalize(const int* __restrict__ perms,
                           const int* __restrict__ target,
                           const float* __restrict__ logits,
                           const float* __restrict__ rowLogZ,
                           const unsigned long long* __restrict__ best,
                           float* __restrict__ out) {
    const unsigned best_idx = 0xFFFFFFFFu - (unsigned)(*best & 0xFFFFFFFFull);
    const int j = threadIdx.x;
    if (j < M_ROWS) {
        const int pj = perms[(size_t)best_idx * 10 + j];
        const int tb = target[pj];
        out[j]          = rowLogZ[j] - logits[(size_t)j * VOCAB + tb];
        out[M_ROWS + j] = (float)tb;
    }
}

extern "C" void kernel_launch(void* const* d_in, const int* in_sizes, int n_in,
                              void* d_out, int out_size, void* d_ws, size_t ws_size,
                              hipStream_t stream) {
    const float* logits = (const float*)d_in[0];   // 10 * 50257 f32
    const int*   target = (const int*)d_in[1];     // 10 i32
    const int*   perms  = (const int*)d_in[2];     // 3628800 * 10 i32
    float*       out    = (float*)d_out;           // 20 f32: loss[10], target_best[10]

    char* ws = (char*)d_ws;
    float* rowLogZ            = (float*)(ws + 0);     // 16 f32
    float* Sbuf               = (float*)(ws + 64);    // 100 f32
    unsigned long long* best  = (unsigned long long*)(ws + 512);

    k_rowstats<<<M_ROWS, 256, 0, stream>>>(logits, rowLogZ);
    k_build_S <<<1, 128, 0, stream>>>(logits, target, rowLogZ, Sbuf, best);
    k_score   <<<NPERM / SCORE_BLOCK, SCORE_BLOCK, 0, stream>>>(perms, Sbuf, best);
    k_finalize<<<1, 32, 0, stream>>>(perms, target, logits, rowLogZ, best, out);
}